// NeuralKNN_16338055594575
// MI455X (gfx1250) — compile-verified
//
#include <hip/hip_runtime.h>
#include <hip/hip_bf16.h>
#include <math.h>

#define CAPACITY   100000
#define KEY_DIM    128
#define KNN        32
#define EPS        1e-8f
#define NQ         2048
#define QTILE      16
#define CHUNK      128
#define NCHUNK     ((CAPACITY + CHUNK - 1) / CHUNK)   // 782
#define LDS_STRIDE 132                                 // padded row stride (floats)
#define QCAP       128
#define KEYBUF_BYTES (CHUNK * LDS_STRIDE * 4)          // 67584

typedef float v2f __attribute__((ext_vector_type(2)));
typedef float v4f __attribute__((ext_vector_type(4)));
typedef float v8f __attribute__((ext_vector_type(8)));
typedef unsigned int v4u __attribute__((ext_vector_type(4)));
typedef int v4i __attribute__((ext_vector_type(4)));
typedef int v8i __attribute__((ext_vector_type(8)));

// Dynamic LDS layout (byte offsets — dynamic segment assumed to start at LDS 0,
// no static __shared__ in this kernel):
//   bufA     @ 0          : 67584
//   bufB     @ 67584      : 67584
//   qd       @ 135168     : QTILE*QCAP*4 = 8192
//   qi       @ 143360     : 8192
//   list_d   @ 151552     : 2048
//   list_i   @ 153600     : 2048
//   tau/qcnt/q2s          : 192
#define SMEM_BYTES (2*KEYBUF_BYTES + QTILE*QCAP*8 + QTILE*KNN*8 + QTILE*12)

// ---------------------------------------------------------------------------
// Pass 0: per-key squared norms. One wave per key (wave32), b128 loads.
// ---------------------------------------------------------------------------
__global__ void knn_keynorms_kernel(const float* __restrict__ keys,
                                    float* __restrict__ k2) {
  const int wave = threadIdx.x >> 5;
  const int lane = threadIdx.x & 31;
  const int key  = blockIdx.x * 8 + wave;
  if (key >= CAPACITY) return;
  const v4f* row = (const v4f*)(keys + (size_t)key * KEY_DIM);
  v4f v = row[lane];                       // 32 lanes * 4 floats = 128
  float s = v.x*v.x + v.y*v.y + v.z*v.z + v.w*v.w;
  s += __shfl_xor(s, 16, 32);
  s += __shfl_xor(s,  8, 32);
  s += __shfl_xor(s,  4, 32);
  s += __shfl_xor(s,  2, 32);
  s += __shfl_xor(s,  1, 32);
  if (lane == 0) k2[key] = s;
}

// ---------------------------------------------------------------------------
// Main kernel: 16 queries per workgroup, 8 waves. Keys stream through a
// double-buffered LDS tile filled by the Tensor Data Mover (TDM) one chunk
// ahead; distances via chained V_WMMA_F32_16X16X4_F32; streaming threshold
// top-32 selection per query.
// ---------------------------------------------------------------------------
__global__ void knn_main_kernel(const float* __restrict__ queries,
                                const float* __restrict__ keys,
                                const float* __restrict__ values,
                                const float* __restrict__ k2g,
                                float* __restrict__ out) {
  extern __shared__ char smem_raw[];
  float* bufA     = (float*)smem_raw;                       // [CHUNK][LDS_STRIDE]
  float* bufB     = bufA + CHUNK * LDS_STRIDE;
  float* qd       = bufB + CHUNK * LDS_STRIDE;              // [QTILE][QCAP]
  int*   qi       = (int*)(qd + QTILE * QCAP);              // [QTILE][QCAP]
  float* list_d   = (float*)(qi + QTILE * QCAP);            // [QTILE][KNN]
  int*   list_i   = (int*)(list_d + QTILE * KNN);           // [QTILE][KNN]
  float* tau      = (float*)(list_i + QTILE * KNN);         // [QTILE]
  int*   qcnt     = (int*)(tau + QTILE);                    // [QTILE]
  float* q2s      = (float*)(qcnt + QTILE);                 // [QTILE]

  const int tid   = threadIdx.x;
  const int wave  = tid >> 5;
  const int lane  = tid & 31;
  const int half  = lane >> 4;       // 0: lanes 0-15, 1: lanes 16-31
  const int l16   = lane & 15;
  const int qbase = blockIdx.x * QTILE;

  // --- TDM issue: 2D tile CHUNKxKEY_DIM f32, LDS rows padded to LDS_STRIDE ---
  // D# built per CDNA5 ISA ch.8: group0 {count,lds_addr,global_addr,type=2},
  // group1 {data_size=4B, pad 4 DWORDs every 128 DWORDs, dims/strides/tile}.
  auto issue_tdm = [&](int chunk, unsigned lds_byte_off) {
    unsigned long long gaddr =
        (unsigned long long)(uintptr_t)keys +
        (unsigned long long)chunk * (CHUNK * KEY_DIM * 4);
    unsigned rows_left = (unsigned)(CAPACITY - chunk * CHUNK); // > 0 always
    v4u g0;
    g0.x = 0x1u;                                        // count=1, user mode
    g0.y = lds_byte_off;                                // lds_addr
    g0.z = (unsigned)(gaddr & 0xFFFFFFFFu);             // global_addr[31:0]
    g0.w = (unsigned)((gaddr >> 32) & 0x01FFFFFFu)      // global_addr[56:32]
           | (2u << 30);                                // type = 2 (image)
    v8i g1;
    g1[0] = (2 << 16)        // data_size = 4 bytes
          | (1 << 20)        // pad_enable
          | (6 << 22)        // pad_interval: every 128 DWORDs
          | (3 << 25);       // pad_amount: 4 DWORDs  -> row stride 132 floats
    g1[1] = (KEY_DIM & 0xFFFF) << 16;                   // tensor_dim0 = 128 (lo16)
    g1[2] = (int)((rows_left & 0xFFFFu) << 16);         // dim0 hi=0 | tensor_dim1 lo16
    g1[3] = (int)(((rows_left >> 16) & 0xFFFFu)         // tensor_dim1 hi16
           | ((unsigned)CHUNK << 16));                  // tile_dim0 = 128
    g1[4] = CHUNK;                                      // tile_dim1 = 128, tile_dim2 = 0
    g1[5] = KEY_DIM;                                    // tensor_dim0_stride lo32 = 128
    g1[6] = 0;                                          // stride hi | dim1_stride lo
    g1[7] = 0;                                          // dim1_stride hi
    v4i z4 = (v4i){0, 0, 0, 0};                         // groups 2/3 unused (2D)
    v8i z8 = (v8i){0, 0, 0, 0, 0, 0, 0, 0};             // unused trailing group
    __builtin_amdgcn_tensor_load_to_lds(g0, g1, z4, z4, z8, 0);
  };

  // Prefetch chunk 0 into bufA while we set up.
  if (wave == 0) issue_tdm(0, 0u);

  // ---- Preload A (queries tile 16x128, f32 WMMA 16x16x4 A layout) ----
  v2f a[32];
  {
    const float* qrow = queries + (size_t)(qbase + l16) * KEY_DIM + half * 2;
#pragma unroll
    for (int s = 0; s < 32; ++s)
      a[s] = *(const v2f*)(qrow + s * 4);
  }

  // q^2 per query from the A registers
  float q2p = 0.f;
#pragma unroll
  for (int s = 0; s < 32; ++s) q2p += a[s].x * a[s].x + a[s].y * a[s].y;
  q2p += __shfl_xor(q2p, 16, 32);
  if (wave == 0 && lane < QTILE) q2s[lane] = q2p;

  // init selection state
  for (int i = tid; i < QTILE * KNN; i += 256) {
    list_d[i] = __builtin_inff();
    list_i[i] = 0;
  }
  if (tid < QTILE) { tau[tid] = __builtin_inff(); qcnt[tid] = 0; }

  if (wave == 0) __builtin_amdgcn_s_wait_tensorcnt(0);  // chunk 0 landed
  __syncthreads();

  float q2v[8];
#pragma unroll
  for (int r = 0; r < 8; ++r) q2v[r] = q2s[r + 8 * half];

  // serial merge of queued candidates into the per-query top-32 list
  auto do_insert = [&](int q) {
    int cnt = qcnt[q]; if (cnt > QCAP) cnt = QCAP;
    float t = tau[q];
    for (int j = 0; j < cnt; ++j) {
      float d = qd[q * QCAP + j];
      if (d < t) {
        int id = qi[q * QCAP + j];
        int mp = 0; float mv = list_d[q * KNN];
        for (int e = 1; e < KNN; ++e) {
          float v = list_d[q * KNN + e];
          if (v > mv) { mv = v; mp = e; }
        }
        list_d[q * KNN + mp] = d;
        list_i[q * KNN + mp] = id;
        mv = list_d[q * KNN];
        for (int e = 1; e < KNN; ++e) mv = fmaxf(mv, list_d[q * KNN + e]);
        t = mv;
      }
    }
    qcnt[q] = 0;
    tau[q]  = t;
  };

  for (int c = 0; c < NCHUNK; ++c) {
    // wave 0: start DMA of next chunk into the buffer we just freed
    if (wave == 0 && (c + 1) < NCHUNK)
      issue_tdm(c + 1, ((c + 1) & 1) ? (unsigned)KEYBUF_BYTES : 0u);

    // wave 1: merge previous chunk's queued candidates (serial per query)
    if (tid >= 32 && tid < 32 + QTILE) do_insert(tid - 32);
    __syncthreads();   // tau updated, queues reset

    const float* kb = (c & 1) ? bufB : bufA;

    float tv[8];
#pragma unroll
    for (int r = 0; r < 8; ++r) tv[r] = tau[r + 8 * half];

    const int keyidx = c * CHUNK + wave * 16 + l16;
    const float k2v = (keyidx < CAPACITY) ? k2g[keyidx] : __builtin_inff();

    // ---- 16x16 dot-product tile via 32 chained f32 WMMAs (K=128) ----
    v8f acc = {};
    const float* brow = kb + (wave * 16 + l16) * LDS_STRIDE + half * 2;
#pragma unroll
    for (int s = 0; s < 32; ++s) {
      v2f b = *(const v2f*)(brow + s * 4);
      acc = __builtin_amdgcn_wmma_f32_16x16x4_f32(
          /*neg_a=*/false, a[s], /*neg_b=*/false, b,
          /*c_mod=*/(short)0, acc, /*reuse_a=*/false, /*reuse_b=*/false);
    }

    // ---- distance + threshold filter + queue append ----
#pragma unroll
    for (int r = 0; r < 8; ++r) {
      float d2 = q2v[r] + k2v - 2.0f * acc[r];
      float d  = sqrtf(fmaxf(d2, 1e-12f));
      if (keyidx < CAPACITY && d < tv[r]) {
        int q   = r + 8 * half;
        int pos = atomicAdd(&qcnt[q], 1);
        if (pos < QCAP) { qd[q * QCAP + pos] = d; qi[q * QCAP + pos] = keyidx; }
      }
    }

    // next chunk's DMA must have landed before anyone reads its buffer
    if (wave == 0) __builtin_amdgcn_s_wait_tensorcnt(0);
    __syncthreads();   // buffer handoff + queue writes visible
  }

  // final merge + output
  if (tid >= 32 && tid < 32 + QTILE) do_insert(tid - 32);
  __syncthreads();

  if (tid < QTILE) {
    float s = 0.f;
    for (int e = 0; e < KNN; ++e) {
      float d = list_d[tid * KNN + e];
      s += (d / (d + EPS)) * values[list_i[tid * KNN + e]];
    }
    out[qbase + tid] = s * (1.0f / KNN);
  }
}

// ---------------------------------------------------------------------------
extern "C" void kernel_launch(void* const* d_in, const int* in_sizes, int n_in,
                              void* d_out, int out_size, void* d_ws, size_t ws_size,
                              hipStream_t stream) {
  const float* queries = (const float*)d_in[0];   // (2048, 128) f32
  const float* keys    = (const float*)d_in[1];   // (100000, 128) f32
  const float* values  = (const float*)d_in[2];   // (100000, 1) f32
  float* k2  = (float*)d_ws;                      // 100000 floats of scratch
  float* out = (float*)d_out;                     // (2048, 1) f32

  knn_keynorms_kernel<<<(CAPACITY + 7) / 8, 256, 0, stream>>>(keys, k2);
  knn_main_kernel<<<NQ / QTILE, 256, SMEM_BYTES, stream>>>(queries, keys, values, k2, out);
}